// GCN_80977313399676
// MI455X (gfx1250) — compile-verified
//
#include <hip/hip_runtime.h>
#include <hip/hip_bf16.h>

typedef __attribute__((ext_vector_type(2))) float v2f;
typedef __attribute__((ext_vector_type(4))) float v4f;
typedef __attribute__((ext_vector_type(8))) float v8f;
typedef __attribute__((ext_vector_type(4))) int   v4i;

// Async global->LDS staging (CDNA5 GLOBAL_LOAD_ASYNC_TO_LDS_B128), with a
// safe synchronous fallback if the builtin is not declared by this toolchain.
#if defined(__gfx1250__) && __has_builtin(__builtin_amdgcn_global_load_async_to_lds_b128)
#define GCN_ASYNC_LDS 1
#else
#define GCN_ASYNC_LDS 0
#endif

typedef __attribute__((address_space(1))) v4i* gcn_gptr_b128;   // global b128
typedef __attribute__((address_space(3))) v4i* gcn_lptr_b128;   // LDS b128

__device__ __forceinline__ void gcn_wait_asynccnt0() {
#if __has_builtin(__builtin_amdgcn_s_wait_asynccnt)
    __builtin_amdgcn_s_wait_asynccnt(0);
#else
    asm volatile("s_wait_asynccnt 0x0" ::: "memory");
#endif
}

__device__ __forceinline__ void gcn_copy_b128_to_lds(const float* src, float* lds_dst) {
#if GCN_ASYNC_LDS
    __builtin_amdgcn_global_load_async_to_lds_b128(
        (gcn_gptr_b128)const_cast<float*>(src),
        (gcn_lptr_b128)lds_dst,
        /*offset=*/0, /*cpol=*/0);
#else
    *(v4f*)lds_dst = *(const v4f*)src;
#endif
}

// ---------------------------------------------------------------------------
// Degree / normalization kernels
// ---------------------------------------------------------------------------
__global__ void gcn_deg_init(float* __restrict__ deg, int N) {
    int i = blockIdx.x * blockDim.x + threadIdx.x;
    if (i < N) deg[i] = 1.0f;   // self loop
}

__global__ void gcn_deg_edges(const int* __restrict__ cols, float* __restrict__ deg, int E) {
    int e = blockIdx.x * blockDim.x + threadIdx.x;
    if (e < E) unsafeAtomicAdd(&deg[cols[e]], 1.0f);
}

__global__ void gcn_deg_to_dinv(float* __restrict__ deg, int N) {
    int i = blockIdx.x * blockDim.x + threadIdx.x;
    if (i < N) deg[i] = rsqrtf(deg[i]);   // deg >= 1 always
}

// ---------------------------------------------------------------------------
// WMMA fp32 GEMM: H[N,MOUT] = (RELU? relu(A) : A)[N,KDIM] @ W[KDIM,MOUT]
// A-tile staged (async) into padded LDS once per block; one wave computes a
// 16x16 tile via V_WMMA_F32_16X16X4_F32 (K-step 4). 256 threads = 8 waves.
// ---------------------------------------------------------------------------
template<int KDIM, int MOUT, bool RELU>
__global__ __launch_bounds__(256) void gcn_gemm_wmma(const float* __restrict__ A,
                                                     const float* __restrict__ W,
                                                     float* __restrict__ H, int N) {
    constexpr int COLT = MOUT / 16;   // column tiles across the 8 waves
    constexpr int ROWT = 8 / COLT;    // row tiles per block
    constexpr int ROWS = ROWT * 16;   // rows staged per block (16 or 32)
    constexpr int LDA  = KDIM + 4;    // pad: stride%64==4 -> conflict-free b64 reads
    __shared__ float sA[ROWS * LDA];

    const int tid    = threadIdx.x;
    const int rowBlk = blockIdx.x * ROWS;

    // ---- cooperative stage of A tile (ROWS x KDIM, row-contiguous) ------
    constexpr int CHR = KDIM / 4;     // b128 chunks per row
    constexpr int NCH = ROWS * CHR;   // 512 or 1024 chunks; 256 threads
#pragma unroll
    for (int c = tid; c < NCH; c += 256) {
        const int rr = c / CHR;
        const int cc = (c % CHR) * 4;
        int gr = rowBlk + rr; if (gr >= N) gr = N - 1;   // clamp (tail block)
        gcn_copy_b128_to_lds(A + (size_t)gr * KDIM + cc, &sA[rr * LDA + cc]);
    }
#if GCN_ASYNC_LDS
    gcn_wait_asynccnt0();
#endif
    __syncthreads();

    // ---- WMMA main loop -------------------------------------------------
    const int wave = tid >> 5;
    const int lane = tid & 31;
    const int half = lane >> 4;       // 0: lanes 0-15, 1: lanes 16-31
    const int l15  = lane & 15;
    const int ct   = wave % COLT;
    const int rt   = wave / COLT;
    const int col  = ct * 16 + l15;   // B-frag / C-frag column
    const float* aRow = &sA[(rt * 16 + l15) * LDA];

    v8f acc = {0.f, 0.f, 0.f, 0.f, 0.f, 0.f, 0.f, 0.f};

#pragma unroll 8
    for (int k = 0; k < KDIM; k += 4) {
        const int ka = k + 2 * half;              // this lane's K pair
        v2f a = *(const v2f*)(aRow + ka);         // A: M=l15, K={ka,ka+1} (LDS)
        if (RELU) {
            a.x = fmaxf(a.x, 0.f);
            a.y = fmaxf(a.y, 0.f);
        }
        v2f b;                                    // B: N=col, K={ka,ka+1}
        b.x = W[(size_t)ka * MOUT + col];
        b.y = W[(size_t)(ka + 1) * MOUT + col];
        acc = __builtin_amdgcn_wmma_f32_16x16x4_f32(
            /*neg_a=*/false, a, /*neg_b=*/false, b,
            /*c_mod=*/(short)0, acc, /*reuse_a=*/false, /*reuse_b=*/false);
    }

    // ---- epilogue: C/D VGPR j -> row (j + 8*half) ----------------------
    const int rowBase = rowBlk + rt * 16;
    if (rowBase + 15 < N) {                       // fast path: whole tile in range
#pragma unroll
        for (int j = 0; j < 8; ++j)
            H[(size_t)(rowBase + j + 8 * half) * MOUT + col] = acc[j];
    } else {
#pragma unroll
        for (int j = 0; j < 8; ++j) {
            const int r = rowBase + j + 8 * half;
            if (r < N) H[(size_t)r * MOUT + col] = acc[j];
        }
    }
}

// ---------------------------------------------------------------------------
// Aggregation init: out[i,f] = h[i,f]*dinv[i]^2 (self loop) + bias[f]
// ---------------------------------------------------------------------------
template<int F>
__global__ void gcn_agg_init(const float* __restrict__ h, const float* __restrict__ dinv,
                             const float* __restrict__ bias, float* __restrict__ out, int N) {
    size_t i = (size_t)blockIdx.x * blockDim.x + threadIdx.x;
    if (i >= (size_t)N * F) return;
    const int node = (int)(i / F);
    const int f    = (int)(i % F);
    const float d  = dinv[node];
    out[i] = h[i] * d * d + bias[f];
}

// ---------------------------------------------------------------------------
// Edge scatter: out[col] += h[row] * dinv[row]*dinv[col]; one wave per edge,
// each lane owns F/32 contiguous floats: one vector gather + native
// global_atomic_add_f32 scatters.
// ---------------------------------------------------------------------------
template<int F>
__global__ __launch_bounds__(256) void gcn_agg_edges(const float* __restrict__ h,
                                                     const float* __restrict__ dinv,
                                                     const int* __restrict__ rows,
                                                     const int* __restrict__ cols,
                                                     float* __restrict__ out, int E) {
    constexpr int VPT = F / 32;   // floats per lane (4 or 2)
    const int lane = threadIdx.x & 31;
    const int wid  = (int)(((size_t)blockIdx.x * blockDim.x + threadIdx.x) >> 5);
    if (wid >= E) return;
    const int r = rows[wid];
    const int c = cols[wid];
    const float nrm = dinv[r] * dinv[c];
    const float* hs = h   + (size_t)r * F + lane * VPT;
    float*       od = out + (size_t)c * F + lane * VPT;
    if constexpr (VPT == 4) {
        v4f hv = *(const v4f*)hs;                 // one b128 gather per lane
        unsafeAtomicAdd(od + 0, hv.x * nrm);
        unsafeAtomicAdd(od + 1, hv.y * nrm);
        unsafeAtomicAdd(od + 2, hv.z * nrm);
        unsafeAtomicAdd(od + 3, hv.w * nrm);
    } else {
        v2f hv = *(const v2f*)hs;                 // one b64 gather per lane
        unsafeAtomicAdd(od + 0, hv.x * nrm);
        unsafeAtomicAdd(od + 1, hv.y * nrm);
    }
}

// ---------------------------------------------------------------------------
// Host-side orchestration
// ---------------------------------------------------------------------------
extern "C" void kernel_launch(void* const* d_in, const int* in_sizes, int n_in,
                              void* d_out, int out_size, void* d_ws, size_t ws_size,
                              hipStream_t stream) {
    const float* x  = (const float*)d_in[0];
    const int*   ei = (const int*)  d_in[1];
    const float* W0 = (const float*)d_in[2];
    const float* b0 = (const float*)d_in[3];
    const float* W1 = (const float*)d_in[4];
    const float* b1 = (const float*)d_in[5];
    const float* W2 = (const float*)d_in[6];
    const float* b2 = (const float*)d_in[7];
    float* out = (float*)d_out;

    const int N = in_sizes[0] / 128;   // 100000
    const int E = in_sizes[1] / 2;     // 1600000
    const int* rows = ei;              // edge_index[0] = source
    const int* cols = ei + E;          // edge_index[1] = target

    // Workspace layout: dinv[N] | bufA[N*128] | bufB[N*128]
    char* ws = (char*)d_ws;
    float* dinv = (float*)ws;
    size_t off  = (((size_t)N * sizeof(float)) + 255) & ~(size_t)255;
    float* bufA = (float*)(ws + off);
    float* bufB = (float*)(ws + off + (size_t)N * 128 * sizeof(float));

    const int T = 256;
    const int nodeBlocks  = (N + T - 1) / T;
    const int edgeBlocks  = (E + T - 1) / T;
    const int edgeWaveBlk = (E + 7) / 8;                  // 1 wave per edge
    const int gemmBlocks128 = (N + 15) / 16;              // MOUT=128: 16 rows/block
    const int gemmBlocks64  = (N + 31) / 32;              // MOUT=64 : 32 rows/block
    const int elemBlocks128 = (int)(((size_t)N * 128 + T - 1) / T);
    const int elemBlocks64  = (int)(((size_t)N * 64  + T - 1) / T);

    // --- normalization: deg -> dinv -------------------------------------
    gcn_deg_init   <<<nodeBlocks, T, 0, stream>>>(dinv, N);
    gcn_deg_edges  <<<edgeBlocks, T, 0, stream>>>(cols, dinv, E);
    gcn_deg_to_dinv<<<nodeBlocks, T, 0, stream>>>(dinv, N);

    // --- layer 0: h = relu(agg(x@W0) + b0) ------------------------------
    gcn_gemm_wmma<128, 128, false><<<gemmBlocks128, T, 0, stream>>>(x, W0, bufA, N);
    gcn_agg_init<128> <<<elemBlocks128, T, 0, stream>>>(bufA, dinv, b0, bufB, N);
    gcn_agg_edges<128><<<edgeWaveBlk,   T, 0, stream>>>(bufA, dinv, rows, cols, bufB, E);

    // --- layer 1 (ReLU of layer-0 fused into A-frag read) ---------------
    gcn_gemm_wmma<128, 128, true><<<gemmBlocks128, T, 0, stream>>>(bufB, W1, bufA, N);
    gcn_agg_init<128> <<<elemBlocks128, T, 0, stream>>>(bufA, dinv, b1, bufB, N);
    gcn_agg_edges<128><<<edgeWaveBlk,   T, 0, stream>>>(bufA, dinv, rows, cols, bufB, E);

    // --- layer 2 (ReLU of layer-1 fused; no final ReLU) -----------------
    gcn_gemm_wmma<128, 64, true><<<gemmBlocks64, T, 0, stream>>>(bufB, W2, bufA, N);
    gcn_agg_init<64> <<<elemBlocks64, T, 0, stream>>>(bufA, dinv, b2, out, N);
    gcn_agg_edges<64><<<edgeWaveBlk,  T, 0, stream>>>(bufA, dinv, rows, cols, out, E);
}